// MagnitudeMaxPool3Dt_27625229648243
// MI455X (gfx1250) — compile-verified
//
#include <hip/hip_runtime.h>

// Magnitude-argmax (2,2,2) pooling, x: (2,8,32,128,128,8) f32 -> (2,8,16,64,64,8) f32.
// Memory-bound streaming kernel: 256 MB read + 32 MB write ~= 12.4 us at 23.3 TB/s.
// Each lane owns 4 consecutive channels of one output voxel -> all traffic is b128,
// non-temporal (input is single-touch and larger than the 192 MB L2).

typedef float v4f __attribute__((ext_vector_type(4)));

// Input geometry (in float4 = 16B units):
//   c-group stride : 1        (C=8 -> 2 groups/voxel)
//   w stride       : 2
//   h stride       : 256      (128 * 8 / 4)
//   d stride       : 32768    (128 * 128 * 8 / 4)
//   n stride       : 1048576  (32 * 128*128*8 / 4), n = b*8 + t
#define W_STR 2u
#define H_STR 256u
#define D_STR 32768u
#define N_STR 1048576u

__global__ __launch_bounds__(256)
void magpool3d_b128_nt_kernel(const v4f* __restrict__ x,
                              v4f* __restrict__ out,
                              unsigned total_v4) {
    unsigned g = blockIdx.x * 256u + threadIdx.x;
    if (g >= total_v4) return;

    // Output flat (float4 units): ((((n*16 + d')*64 + h')*64 + w')*2 + cg)
    unsigned cg = g & 1u;
    unsigned wp = (g >> 1) & 63u;
    unsigned hp = (g >> 7) & 63u;
    unsigned dp = (g >> 13) & 15u;
    unsigned n  = g >> 17;

    unsigned base = n * N_STR + (dp << 1) * D_STR + (hp << 1) * H_STR
                  + (wp << 2) /* (2*wp)*W_STR */ + cg;

    // Issue all 8 window loads up front (8-deep MLP per lane), NT hint.
    v4f v[8];
#pragma unroll
    for (int i = 0; i < 2; ++i)
#pragma unroll
        for (int j = 0; j < 2; ++j)
#pragma unroll
            for (int k = 0; k < 2; ++k)
                v[(i << 2) | (j << 1) | k] = __builtin_nontemporal_load(
                    x + base + (unsigned)i * D_STR + (unsigned)j * H_STR
                             + (unsigned)k * W_STR);

    // First-max tie-breaking (jnp.argmax): window order i*4 + j*2 + k,
    // replace only on strictly larger |v|.
    float bv[4], ba[4];
#pragma unroll
    for (int s = 0; s < 4; ++s) {
        bv[s] = v[0][s];
        ba[s] = __builtin_fabsf(v[0][s]);
    }
#pragma unroll
    for (int w = 1; w < 8; ++w) {
#pragma unroll
        for (int s = 0; s < 4; ++s) {
            float val = v[w][s];
            float a   = __builtin_fabsf(val);
            if (a > ba[s]) { ba[s] = a; bv[s] = val; }
        }
    }

    v4f o = {bv[0], bv[1], bv[2], bv[3]};
    __builtin_nontemporal_store(o, out + g);
}

extern "C" void kernel_launch(void* const* d_in, const int* in_sizes, int n_in,
                              void* d_out, int out_size, void* d_ws, size_t ws_size,
                              hipStream_t stream) {
    (void)in_sizes; (void)n_in; (void)d_ws; (void)ws_size;
    const v4f* x = (const v4f*)d_in[0];
    v4f* out     = (v4f*)d_out;

    // out_size = 2*8*16*64*64*8 = 8,388,608 floats -> 2,097,152 float4 lanes.
    unsigned total_v4 = (unsigned)(out_size / 4);
    unsigned blocks   = (total_v4 + 255u) / 256u;   // 8192 blocks of 256 (8 waves/WG)

    magpool3d_b128_nt_kernel<<<blocks, 256, 0, stream>>>(x, out, total_v4);
}